// Step3_29703993819496
// MI455X (gfx1250) — compile-verified
//
#include <hip/hip_runtime.h>
#include <math.h>

// ---------------------------------------------------------------------------
// out = irfft( rfft(log(mirror(x))).real * sm[f0] * cp[f0] )[:1025]
// collapses to two fp32 GEMMs against the symmetric cosine matrix
// D[k][n] = cos(pi*k*n/1024):
//   y[f,n] = a_n * log x[f,n]                     (a_0 = a_1024 = 1, else 2)
//   C      = y @ D
//   L[f,k] = C[f,k] * sm[f0[f],k] * cp[f0[f],k] * a_k / 2048
//   out    = (L @ D)[:, :1025]
// Implemented with V_WMMA_F32_16X16X4_F32 (full-precision matrix path).
// Wave tile: 32x64 (2 M-tiles x 4 N-tiles) -> 8 WMMA per 10 loads.
// ---------------------------------------------------------------------------

typedef float v2f __attribute__((ext_vector_type(2)));
typedef float v8f __attribute__((ext_vector_type(8)));

#define BIN       1025
#define NPAD      1088            // 17 * 64, padded N/K dimension
#define M_FRAMES  32000           // B*T = 16*2000
#define KITER     257             // k = 0..1027 in steps of 4 (cols >=1025 are 0)
#define NT64      17              // N tiles of 64 -> 1088
#define MT32      1000            // M tiles of 32 -> 32000
#define PI_OVER_1024 3.067961575771282e-3f   // pi/1024

// ---------------- prep: cosine matrix, padded with zeros --------------------
__global__ __launch_bounds__(256) void prep_cos(float* __restrict__ D) {
    int idx = blockIdx.x * 256 + threadIdx.x;
    if (idx >= NPAD * NPAD) return;
    int r = idx / NPAD;
    int c = idx % NPAD;
    float v = 0.0f;
    if (r < BIN && c < BIN) {
        int m = (r * c) & 2047;            // cos(pi*m/1024) has period 2048 in m
        v = cosf((float)m * PI_OVER_1024);
    }
    D[idx] = v;
}

// ---------------- prep: y = a_n * log(x), zero-padded -----------------------
__global__ __launch_bounds__(256) void prep_logx(const float* __restrict__ x,
                                                 float* __restrict__ y) {
    long long idx = (long long)blockIdx.x * 256 + threadIdx.x;
    if (idx >= (long long)M_FRAMES * NPAD) return;
    int f = (int)(idx / NPAD);
    int n = (int)(idx % NPAD);
    float v = 0.0f;
    if (n < BIN) {
        float a = (n == 0 || n == BIN - 1) ? 1.0f : 2.0f;
        v = a * __logf(x[(long long)f * BIN + n]);
    }
    y[idx] = v;
}

// ---------------- WMMA GEMM: 32x64 tile per wave, K step 4 ------------------
// MODE 0: Out = lifter-epilogue( A @ Bm )  -> L  (NPAD-stride, zero pad cols)
// MODE 1: Out = ( A @ Bm )[:, :BIN]        -> final output (BIN-stride)
template <int MODE>
__global__ __launch_bounds__(256) void gemm_wmma(
    const float* __restrict__ A,   // [M_FRAMES x NPAD] row-major
    const float* __restrict__ Bm,  // [NPAD x NPAD] cosine matrix
    float* __restrict__ Out,
    const int* __restrict__ f0,    // [M_FRAMES] (MODE 0)
    const float* __restrict__ sm,  // [801 x BIN]       (MODE 0)
    const float* __restrict__ cp)  // [801 x BIN]       (MODE 0)
{
    const int lane = threadIdx.x & 31;
    const int wave = threadIdx.x >> 5;
    const int tile = blockIdx.x * 8 + wave;      // 1000*17 = 17000 wave tiles
    // Consecutive waves in a block share the SAME B tile (identical fragment
    // addresses -> L0/WGP$ hits on the hot cosine matrix) and walk contiguous
    // A row-blocks. 1000 % 8 == 0, so a block never straddles an nT boundary.
    const int mT   = tile % MT32;
    const int nT   = tile / MT32;
    const int mBase = mT * 32;
    const int nBase = nT * 64;
    const int half  = lane >> 4;                  // 0: lanes 0-15, 1: lanes 16-31
    const int l16   = lane & 15;

    v8f acc[2][4];
#pragma unroll
    for (int mi = 0; mi < 2; ++mi)
#pragma unroll
        for (int j = 0; j < 4; ++j) acc[mi][j] = (v8f){0,0,0,0,0,0,0,0};

    // A fragment addr: lane l (half h) holds A[row][k + 2h], [k + 2h + 1]
    const float* aRow0 = A + (long long)(mBase + l16) * NPAD + 2 * half;
    const float* aRow1 = aRow0 + (long long)16 * NPAD;
    // B fragment addr: vgpr0 = B[k+half][col], vgpr1 = B[k+2+half][col]
    const float* bCol = Bm + (long long)half * NPAD + nBase + l16;

    // unroll 1: keep one accumulator register set so the allocator ties
    // wmma vdst == src2 (no v_mov ping-pong between unrolled bodies).
#pragma unroll 1
    for (int i = 0; i < KITER; ++i) {
        const int k = i * 4;
        v2f a0 = *(const v2f*)(aRow0 + k);
        v2f a1 = *(const v2f*)(aRow1 + k);
        const float* bp = bCol + (long long)k * NPAD;
        v2f b[4];
#pragma unroll
        for (int j = 0; j < 4; ++j) {
            b[j].x = bp[j * 16];
            b[j].y = bp[2 * NPAD + j * 16];
        }
#pragma unroll
        for (int j = 0; j < 4; ++j) {
            acc[0][j] = __builtin_amdgcn_wmma_f32_16x16x4_f32(
                false, a0, false, b[j], (short)0, acc[0][j], false, false);
            acc[1][j] = __builtin_amdgcn_wmma_f32_16x16x4_f32(
                false, a1, false, b[j], (short)0, acc[1][j], false, false);
        }
    }

    // C/D layout: vgpr v, lanes 0-15 -> (M=v, N=lane); lanes 16-31 -> (M=v+8)
#pragma unroll
    for (int mi = 0; mi < 2; ++mi) {
#pragma unroll
        for (int j = 0; j < 4; ++j) {
            const int n = nBase + j * 16 + l16;       // output column
            if (MODE == 0) {
                if (n < BIN) {
                    const float wk = (n == 0 || n == BIN - 1) ? 1.0f : 2.0f;
                    const float scale = wk * (1.0f / 2048.0f);
#pragma unroll
                    for (int v = 0; v < 8; ++v) {
                        const int f = mBase + mi * 16 + v + 8 * half;
                        const int ff = f0[f];
                        const float lif = sm[(long long)ff * BIN + n] *
                                          cp[(long long)ff * BIN + n];
                        Out[(long long)f * NPAD + n] = acc[mi][j][v] * lif * scale;
                    }
                } else {
#pragma unroll
                    for (int v = 0; v < 8; ++v) {
                        const int f = mBase + mi * 16 + v + 8 * half;
                        Out[(long long)f * NPAD + n] = 0.0f;
                    }
                }
            } else {
                if (n < BIN) {
#pragma unroll
                    for (int v = 0; v < 8; ++v) {
                        const int f = mBase + mi * 16 + v + 8 * half;
                        Out[(long long)f * BIN + n] = acc[mi][j][v];
                    }
                }
            }
        }
    }
}

// ---------------------------------------------------------------------------
extern "C" void kernel_launch(void* const* d_in, const int* in_sizes, int n_in,
                              void* d_out, int out_size, void* d_ws, size_t ws_size,
                              hipStream_t stream) {
    const float* x  = (const float*)d_in[0];   // [16,2000,1025]
    const int*   f0 = (const int*)d_in[1];     // [16,2000]
    const float* sm = (const float*)d_in[2];   // [801,1025]
    const float* cp = (const float*)d_in[3];   // [801,1025]
    float* out = (float*)d_out;                // [16,2000,1025]

    char* ws = (char*)d_ws;
    float* D = (float*)ws;                                       // 1088*1088 f32
    size_t offD = (size_t)NPAD * NPAD * sizeof(float);
    float* y = (float*)(ws + offD);                              // 32000*1088 f32
    size_t offY = (size_t)M_FRAMES * NPAD * sizeof(float);
    float* L = (float*)(ws + offD + offY);                       // 32000*1088 f32

    {   // cosine matrix
        int n = NPAD * NPAD;
        prep_cos<<<(n + 255) / 256, 256, 0, stream>>>(D);
    }
    {   // weighted log spectrum
        long long n = (long long)M_FRAMES * NPAD;
        prep_logx<<<(unsigned)((n + 255) / 256), 256, 0, stream>>>(x, y);
    }
    const int blocks = MT32 * NT64 / 8;   // 2125, exact
    gemm_wmma<0><<<blocks, 256, 0, stream>>>(y, D, L, f0, sm, cp);
    gemm_wmma<1><<<blocks, 256, 0, stream>>>(L, D, out, nullptr, nullptr, nullptr);
}